// G_MTGNN_16423954940301
// MI455X (gfx1250) — compile-verified
//
#include <hip/hip_runtime.h>
#include <math.h>

#define NN   8192
#define DIM  64
#define TOPK 32
#define ALPHA 3.0f
#define ROWS 8          // rows per workgroup in the score kernel

typedef __attribute__((ext_vector_type(2))) float v2f;
typedef __attribute__((ext_vector_type(8))) float v8f;

// fast tanh(3a) for a > 0:  tanh(x) = 1 - 2/(exp(2x)+1); exp overflow -> 1.0
__device__ __forceinline__ float tanh3_pos(float a) {
    return 1.0f - 2.0f * __builtin_amdgcn_rcpf(__expf(6.0f * a) + 1.0f);
}

// -------------------------------------------------------------------------
// Kernel 1: node features  n{1,2}[i][d] = tanh(ALPHA*(emb[idx[i]] . w[d] + b[d]))
// Each thread computes a K-pair (d0, d0+1) so it can emit the WMMA-ready
// packed B layout:  n?P[dd][j] = { n?[j][2*dd], n?[j][2*dd+1] }  (float2),
// letting the score kernel load each B fragment with one global_load_b64.
// -------------------------------------------------------------------------
__global__ __launch_bounds__(256)
void nodes_kernel(const int* __restrict__ idx,
                  const float* __restrict__ emb1, const float* __restrict__ emb2,
                  const float* __restrict__ w1,   const float* __restrict__ b1,
                  const float* __restrict__ w2,   const float* __restrict__ b2,
                  float* __restrict__ n1,  float* __restrict__ n2,
                  float2* __restrict__ n1P, float2* __restrict__ n2P)
{
    int t = blockIdx.x * blockDim.x + threadIdx.x;       // [0, NN*32)
    if (t >= NN * (DIM / 2)) return;
    int i  = t >> 5;            // node
    int dd = t & 31;            // K-pair
    int d0 = dd * 2;
    int row = idx[i];
    const float* e1  = emb1 + (size_t)row * DIM;
    const float* e2  = emb2 + (size_t)row * DIM;
    const float* w1a = w1 + (size_t)d0 * DIM;
    const float* w1b = w1a + DIM;
    const float* w2a = w2 + (size_t)d0 * DIM;
    const float* w2b = w2a + DIM;
    float s1a = b1[d0], s1b = b1[d0 + 1];
    float s2a = b2[d0], s2b = b2[d0 + 1];
#pragma unroll
    for (int k = 0; k < DIM; ++k) {
        float x1 = e1[k], x2 = e2[k];
        s1a = fmaf(x1, w1a[k], s1a);
        s1b = fmaf(x1, w1b[k], s1b);
        s2a = fmaf(x2, w2a[k], s2a);
        s2b = fmaf(x2, w2b[k], s2b);
    }
    float2 v1 = make_float2(tanhf(ALPHA * s1a), tanhf(ALPHA * s1b));
    float2 v2 = make_float2(tanhf(ALPHA * s2a), tanhf(ALPHA * s2b));
    ((float2*)n1)[(size_t)i * (DIM / 2) + dd] = v1;      // row-major (A tile feed)
    ((float2*)n2)[(size_t)i * (DIM / 2) + dd] = v2;
    n1P[(size_t)dd * NN + i] = v1;                       // packed-transposed (B feed)
    n2P[(size_t)dd * NN + i] = v2;
}

// -------------------------------------------------------------------------
// Kernel 2: fused score-matrix (WMMA f32 16x16x4) + per-row top-32 radix
// select in 320KB WGP LDS + masked output.
//
// A operand packs [n1 rows i0..i0+7 ; n2 rows i0..i0+7] (16x64).
//   D1 = A x n2_j^T : lane l<16  holds S1[r][l] = n1_i . n2_j in d1[r], r=0..7
//   D2 = A x n1_j^T : lane l+16  holds S2[r][l] = n2_i . n1_j in d2[r], r=0..7
// Full-lane epilogue: lanes 0-15 finish rows 0..3, lanes 16-31 rows 4..7.
// -------------------------------------------------------------------------
__global__ __launch_bounds__(1024)
void score_topk_kernel(const float* __restrict__ n1,   const float* __restrict__ n2,
                       const float2* __restrict__ n1P, const float2* __restrict__ n2P,
                       const float* __restrict__ noise,
                       float* __restrict__ out)
{
    __shared__ unsigned int keys[ROWS][NN];       // 256 KB (CDNA5: 320KB/WG)
    __shared__ float        Atile[16][DIM];       // 4 KB
    __shared__ unsigned int hist[ROWS][256];      // 8 KB
    __shared__ unsigned int s_prefix[ROWS];
    __shared__ unsigned int s_want[ROWS];

    const int tid  = threadIdx.x;
    const int lane = tid & 31;
    const int wave = tid >> 5;                    // 0..31
    const int i0   = blockIdx.x * ROWS;

    // ---- load packed A tile: rows 0..7 = n1, rows 8..15 = n2 (1024 elems) ----
    {
        int rr = tid >> 6;            // 0..15
        int k  = tid & 63;            // 0..63
        Atile[rr][k] = (rr < ROWS) ? n1[(size_t)(i0 + rr) * DIM + k]
                                   : n2[(size_t)(i0 + rr - ROWS) * DIM + k];
    }
    __syncthreads();

    // ---- preload A fragments (constant across all column blocks) ----
    // 16x4 f32 A layout: lanes 0-15: K={k0,k0+1}; lanes 16-31: K={k0+2,k0+3}; M=lane%16
    v2f a_frag[16];
    {
        int m  = lane & 15;
        int ko = (lane < 16) ? 0 : 2;
#pragma unroll
        for (int kk = 0; kk < 16; ++kk) {
            a_frag[kk].x = Atile[m][4 * kk + ko];
            a_frag[kk].y = Atile[m][4 * kk + ko + 1];
        }
    }

    // ---- sweep column blocks: 512 blocks of 16 columns, 16 per wave ----
    const int jn  = lane & 15;                    // column within block
    const int ddo = (lane < 16) ? 0 : 1;          // which K-pair of the 4-K slab
#pragma unroll 1
    for (int bb = 0; bb < 16; ++bb) {
        const int j0 = ((bb << 5) + wave) << 4;   // this wave's 16-col block

        // prefetch this block's noise segments (consumed after the WMMA chain)
        {
            int pr = lane & 7;
            int pc = (lane >> 3) << 2;            // 0,4,8,12
            __builtin_prefetch(noise + (size_t)(i0 + pr) * NN + j0 + pc, 0, 3);
        }

        // ---- batch-issue all 32 B-fragment loads (deep load clause), then
        //      run the serial WMMA accumulation chain with incremental waits ----
        float2 p1s[16], p2s[16];
#pragma unroll
        for (int kk = 0; kk < 16; ++kk) {
            const size_t poff = (size_t)(2 * kk + ddo) * NN + j0 + jn;
            p1s[kk] = n2P[poff];                   // one global_load_b64 each
            p2s[kk] = n1P[poff];
        }

        v8f d1 = {};
        v8f d2 = {};
#pragma unroll
        for (int kk = 0; kk < 16; ++kk) {
            v2f bf1, bf2;
            bf1.x = p1s[kk].x; bf1.y = p1s[kk].y;
            bf2.x = p2s[kk].x; bf2.y = p2s[kk].y;
            d1 = __builtin_amdgcn_wmma_f32_16x16x4_f32(
                    false, a_frag[kk], false, bf1, (short)0, d1, false, false);
            d2 = __builtin_amdgcn_wmma_f32_16x16x4_f32(
                    false, a_frag[kk], false, bf2, (short)0, d2, false, false);
        }

        // full-lane epilogue: lanes<16 do rows rp, lanes>=16 do rows rp+4
#pragma unroll
        for (int rp = 0; rp < 4; ++rp) {
            float s2lo = __shfl_xor(d2[rp], 16);       // lanes<16: S2[rp][l]
            float s1hi = __shfl_xor(d1[rp + 4], 16);   // lanes>=16: S1[rp+4][l-16]
            const bool lo = (lane < 16);
            float aval = lo ? (d1[rp] - s2lo) : (s1hi - d2[rp + 4]);
            int   r    = lo ? rp : rp + 4;
            float adj  = (aval > 0.0f) ? tanh3_pos(aval) : 0.0f;
            size_t off = (size_t)(i0 + r) * NN + j0 + jn;
            out[off] = adj;                            // stage unmasked adj (L2-hot)
            float nz  = __builtin_nontemporal_load(noise + off);  // single-use stream
            float key = adj + 0.01f * nz;
            unsigned int u = __float_as_uint(key);
            u = (u & 0x80000000u) ? ~u : (u | 0x80000000u);  // order-preserving key
            keys[r][j0 + jn] = u;
        }
    }

    __threadfence();               // make staged adj visible to all waves in WG
    __syncthreads();

    // ---- per-row exact top-32 threshold: 4-pass MSB radix select in LDS ----
    const int r  = tid >> 7;       // 128 threads per row
    const int tt = tid & 127;
    if (tt == 0) { s_prefix[r] = 0u; s_want[r] = TOPK; }
    __syncthreads();

#pragma unroll 1
    for (int pass = 0; pass < 4; ++pass) {
        const int shift = 24 - 8 * pass;
        for (int b = tt; b < 256; b += 128) hist[r][b] = 0u;
        __syncthreads();

        const unsigned int himask = (unsigned int)(0xFFFFFFFFull << (shift + 8));
        const unsigned int P      = s_prefix[r];
        for (int j = tt; j < NN; j += 128) {
            unsigned int u = keys[r][j];
            if ((u & himask) == P)
                atomicAdd(&hist[r][(u >> shift) & 255u], 1u);
        }
        __syncthreads();

        if (tt == 0) {
            unsigned int cum = 0, wantv = s_want[r];
            int bsel = 0;
            for (int b = 255; b >= 0; --b) {
                unsigned int c = hist[r][b];
                if (cum + c >= wantv) { bsel = b; break; }
                cum += c;
            }
            s_prefix[r] |= ((unsigned int)bsel << shift);
            s_want[r]    = wantv - cum;
        }
        __syncthreads();
    }

    // ---- masked rewrite: keep adj where key >= exact K-th largest key ----
    {
        const unsigned int T = s_prefix[r];
        const size_t base = (size_t)(i0 + r) * NN;
        for (int j = tt; j < NN; j += 128) {
            float v = out[base + j];                    // L2-hot readback
            float o = (keys[r][j] >= T) ? v : 0.0f;
            __builtin_nontemporal_store(o, out + base + j);  // write-once stream
        }
    }
}

// -------------------------------------------------------------------------
extern "C" void kernel_launch(void* const* d_in, const int* in_sizes, int n_in,
                              void* d_out, int out_size, void* d_ws, size_t ws_size,
                              hipStream_t stream)
{
    const int*   idx   = (const int*)  d_in[0];
    const float* emb1  = (const float*)d_in[1];
    const float* emb2  = (const float*)d_in[2];
    const float* w1    = (const float*)d_in[3];
    const float* b1    = (const float*)d_in[4];
    const float* w2    = (const float*)d_in[5];
    const float* b2    = (const float*)d_in[6];
    const float* noise = (const float*)d_in[7];
    float* out = (float*)d_out;

    float*  wsf = (float*)d_ws;
    float*  n1  = wsf;                                   // [NN][64]
    float*  n2  = wsf + (size_t)NN * DIM;                // [NN][64]
    float2* n1P = (float2*)(wsf + (size_t)2 * NN * DIM); // [32][NN] packed K-pairs
    float2* n2P = n1P + (size_t)(DIM / 2) * NN;          // [32][NN]

    nodes_kernel<<<(NN * (DIM / 2) + 255) / 256, 256, 0, stream>>>(
        idx, emb1, emb2, w1, b1, w2, b2, n1, n2, n1P, n2P);

    score_topk_kernel<<<NN / ROWS, 1024, 0, stream>>>(
        n1, n2, n1P, n2P, noise, out);
}